// GCN_5214090297939
// MI455X (gfx1250) — compile-verified
//
#include <hip/hip_runtime.h>
#include <math.h>

typedef float v2f __attribute__((ext_vector_type(2)));
typedef float v8f __attribute__((ext_vector_type(8)));

#define GCN_IN 512
#define GCN_MID 256
#define GCN_OUT 40
#define GCN_OUTP 48   // padded to 3 WMMA column tiles

// ---------------- degree / normalization ----------------
__global__ void k_zero_f32(float* __restrict__ p, int n) {
  int i = blockIdx.x * blockDim.x + threadIdx.x;
  if (i < n) p[i] = 0.0f;
}

__global__ void k_deg_scatter(const int* __restrict__ dst, float* __restrict__ deg, int e) {
  int i = blockIdx.x * blockDim.x + threadIdx.x;
  if (i < e) atomicAdd(&deg[dst[i]], 1.0f);
}

__global__ void k_deg_finalize(float* __restrict__ dinv, int n) {
  int i = blockIdx.x * blockDim.x + threadIdx.x;
  if (i < n) dinv[i] = rsqrtf(dinv[i] + 1.0f);  // +1 self loop
}

// ---------------- weight transpose (+ zero pad) : Wt[col*K + k] ----------------
__global__ void k_transpose_pad(const float* __restrict__ W, float* __restrict__ Wt,
                                int K, int M, int Mp) {
  int idx = blockIdx.x * blockDim.x + threadIdx.x;
  if (idx >= K * Mp) return;
  int col = idx / K, k = idx % K;
  Wt[idx] = (col < M) ? W[(size_t)k * M + col] : 0.0f;
}

// ---------------- dense GEMM via WMMA f32 16x16x4 ----------------
// Wave computes a 16(node) x 16*NT(outfeat) tile of H = X @ W.
// A (16x4): lane<16 row lm holds K={ka,ka+1}, lane>=16 holds K={ka+2,ka+3}.
// B (4x16): column = lane&15, same K split; fed from column-major Wt -> b64 loads.
// C/D: 8 VGPRs per tile, lane = column, vgpr j -> row j + 8*(lane>>4).
// blockDim = (32, Mp/(16*NT)); threadIdx.y = column group (shares X rows via WGP$).
template <int K, int Mp, int NT>
__global__ __launch_bounds__(128) void k_gemm_wmma(
    const float* __restrict__ X, const float* __restrict__ Wt,
    float* __restrict__ H) {
  const int lane = threadIdx.x;        // 0..31
  const int half = lane >> 4;
  const int lm   = lane & 15;
  const int tm   = blockIdx.x;                      // node tile (N/16 exact)
  const int colBase = threadIdx.y * (16 * NT);

  const float* xrow = X + (size_t)(tm * 16 + lm) * K;
  v8f acc[NT];
#pragma unroll
  for (int i = 0; i < NT; ++i) acc[i] = (v8f){};

#pragma unroll 8
  for (int k0 = 0; k0 < K; k0 += 4) {
    const int ka = k0 + 2 * half;
    v2f a = *(const v2f*)(xrow + ka);
#pragma unroll
    for (int i = 0; i < NT; ++i) {
      v2f b = *(const v2f*)(Wt + (size_t)(colBase + i * 16 + lm) * K + ka);
      acc[i] = __builtin_amdgcn_wmma_f32_16x16x4_f32(
          false, a, false, b, (short)0, acc[i], false, false);
    }
  }

#pragma unroll
  for (int i = 0; i < NT; ++i) {
#pragma unroll
    for (int j = 0; j < 8; ++j) {
      const int m = tm * 16 + j + 8 * half;
      H[(size_t)m * Mp + colBase + i * 16 + lm] = acc[i][j];
    }
  }
}

// ---------------- aggregation ----------------
// agg[i,f] = h[i,f] * dinv[i]^2 + bias[f]   (H may have padded row stride Hs)
__global__ void k_agg_init(const float* __restrict__ H, const float* __restrict__ dinv,
                           const float* __restrict__ bias, float* __restrict__ A,
                           int n, int M, int Hs) {
  long long idx = (long long)blockIdx.x * blockDim.x + threadIdx.x;
  if (idx >= (long long)n * M) return;
  int i = (int)(idx / M);
  int f = (int)(idx % M);
  float di = dinv[i];
  A[idx] = H[(size_t)i * Hs + f] * di * di + bias[f];
}

// one wave per edge: agg[dst] += h[src] * dinv[src]*dinv[dst]
__global__ __launch_bounds__(256) void k_agg_scatter(
    const float* __restrict__ H, const int* __restrict__ src, const int* __restrict__ dst,
    const float* __restrict__ dinv, float* __restrict__ A, int e, int M, int Hs) {
  int w = blockIdx.x * (blockDim.x >> 5) + (threadIdx.x >> 5);
  int lane = threadIdx.x & 31;
  if (w >= e) return;
  int s = src[w], d = dst[w];
  float nrm = dinv[s] * dinv[d];
  const float* hs = H + (size_t)s * Hs;
  float* ad = A + (size_t)d * M;
  for (int f = lane; f < M; f += 32)
    atomicAdd(&ad[f], hs[f] * nrm);
}

__global__ void k_relu_ip(float* __restrict__ p, long long n) {
  long long i = (long long)blockIdx.x * blockDim.x + threadIdx.x;
  if (i < n) p[i] = fmaxf(p[i], 0.0f);
}

// ---------------- log_softmax over 40 classes, one wave per node ----------------
__global__ __launch_bounds__(256) void k_log_softmax40(
    const float* __restrict__ Z, float* __restrict__ out, int n) {
  int w = blockIdx.x * (blockDim.x >> 5) + (threadIdx.x >> 5);
  int lane = threadIdx.x & 31;
  if (w >= n) return;
  const float* z = Z + (size_t)w * GCN_OUT;
  float v0 = (lane < GCN_OUT) ? z[lane] : -INFINITY;
  float v1 = (lane + 32 < GCN_OUT) ? z[lane + 32] : -INFINITY;
  float mx = fmaxf(v0, v1);
#pragma unroll
  for (int o = 16; o; o >>= 1) mx = fmaxf(mx, __shfl_xor(mx, o, 32));
  float s = ((lane < GCN_OUT) ? __expf(v0 - mx) : 0.0f) +
            ((lane + 32 < GCN_OUT) ? __expf(v1 - mx) : 0.0f);
#pragma unroll
  for (int o = 16; o; o >>= 1) s += __shfl_xor(s, o, 32);
  float lse = mx + __logf(s);
  if (lane < GCN_OUT) out[(size_t)w * GCN_OUT + lane] = v0 - lse;
  if (lane + 32 < GCN_OUT) out[(size_t)w * GCN_OUT + lane + 32] = v1 - lse;
}

// ---------------- host side ----------------
template <int K, int Mp, int NT>
static void run_conv(const float* X, const float* Wt, const float* bias, int Mreal,
                     float* Hbuf, float* Abuf,
                     const int* src, const int* dst, const float* dinv,
                     int n, int e, bool relu, hipStream_t stream) {
  dim3 gblk(32, Mp / (16 * NT), 1);
  dim3 ggrd(n / 16, 1, 1);
  k_gemm_wmma<K, Mp, NT><<<ggrd, gblk, 0, stream>>>(X, Wt, Hbuf);

  long long tot = (long long)n * Mreal;
  k_agg_init<<<(unsigned)((tot + 255) / 256), 256, 0, stream>>>(
      Hbuf, dinv, bias, Abuf, n, Mreal, Mp);

  k_agg_scatter<<<(e + 7) / 8, 256, 0, stream>>>(
      Hbuf, src, dst, dinv, Abuf, e, Mreal, Mp);

  if (relu)
    k_relu_ip<<<(unsigned)((tot + 255) / 256), 256, 0, stream>>>(Abuf, tot);
}

extern "C" void kernel_launch(void* const* d_in, const int* in_sizes, int n_in,
                              void* d_out, int out_size, void* d_ws, size_t ws_size,
                              hipStream_t stream) {
  (void)n_in; (void)out_size; (void)ws_size;
  const float* x   = (const float*)d_in[0];
  const int*  edge = (const int*)d_in[1];
  const float* W1  = (const float*)d_in[2];
  const float* b1  = (const float*)d_in[3];
  const float* Wi  = (const float*)d_in[4];
  const float* bi  = (const float*)d_in[5];
  const float* W2  = (const float*)d_in[6];
  const float* b2  = (const float*)d_in[7];

  const int n = in_sizes[0] / GCN_IN;     // 100000
  const int e = in_sizes[1] / 2;          // 1600000
  const int* src = edge;
  const int* dst = edge + e;

  float* out = (float*)d_out;
  float* emb = out + (size_t)n * GCN_OUT;        // second tuple output [n, 256]

  float* wsf  = (float*)d_ws;
  float* dinv = wsf;                              // n floats
  float* P    = wsf + 100352;                     // n*256 floats (GEMM out)
  float* Q    = P + (size_t)n * GCN_MID;          // n*256 floats (agg out)
  float* Wt1  = Q + (size_t)n * GCN_MID;          // 256*512
  float* Wti  = Wt1 + GCN_MID * GCN_IN;           // 256*256
  float* Wt2  = Wti + GCN_MID * GCN_MID;          // 48*256 (zero padded cols 40..47)

  // 0) transpose/pad weights into column-major for b64 B-operand loads
  k_transpose_pad<<<(GCN_IN * GCN_MID + 255) / 256, 256, 0, stream>>>(
      W1, Wt1, GCN_IN, GCN_MID, GCN_MID);
  k_transpose_pad<<<(GCN_MID * GCN_MID + 255) / 256, 256, 0, stream>>>(
      Wi, Wti, GCN_MID, GCN_MID, GCN_MID);
  k_transpose_pad<<<(GCN_MID * GCN_OUTP + 255) / 256, 256, 0, stream>>>(
      W2, Wt2, GCN_MID, GCN_OUT, GCN_OUTP);

  // 1) normalization: deg -> dinv
  k_zero_f32<<<(n + 255) / 256, 256, 0, stream>>>(dinv, n);
  k_deg_scatter<<<(e + 255) / 256, 256, 0, stream>>>(dst, dinv, e);
  k_deg_finalize<<<(n + 255) / 256, 256, 0, stream>>>(dinv, n);

  // 2) conv1: x(512) -> Q(256), no relu
  run_conv<GCN_IN, GCN_MID, 4>(x, Wt1, b1, GCN_MID, P, Q, src, dst, dinv, n, e, false, stream);
  // 3) conv2, conv3: Q -> Q, relu
  run_conv<GCN_MID, GCN_MID, 4>(Q, Wti, bi, GCN_MID, P, Q, src, dst, dinv, n, e, true, stream);
  run_conv<GCN_MID, GCN_MID, 4>(Q, Wti, bi, GCN_MID, P, Q, src, dst, dinv, n, e, true, stream);
  // 4) conv4: Q -> emb slice of d_out (this IS the `emb` tuple output), relu
  run_conv<GCN_MID, GCN_MID, 4>(Q, Wti, bi, GCN_MID, P, emb, src, dst, dinv, n, e, true, stream);
  // 5) conv5: emb(256) -> Q(40) via padded 48-col GEMM, no relu (relu(emb)==emb)
  run_conv<GCN_MID, GCN_OUTP, 3>(emb, Wt2, b2, GCN_OUT, P, Q, src, dst, dinv, n, e, false, stream);
  // 6) log_softmax over 40 classes into out[0 : n*40]
  k_log_softmax40<<<(n + 7) / 8, 256, 0, stream>>>(Q, out, n);
}